// GATv2Encoder_69750268887410
// MI455X (gfx1250) — compile-verified
//
#include <hip/hip_runtime.h>
#include <math.h>

// Problem constants (match reference)
#define NNODES 10000
#define NEDGES 320000
#define ETOT   (NEDGES + NNODES)
#define FIN    78
#define CCH    128          // channels per head (both layers)
#define H1     4            // heads in layer 1
#define NG     64           // graphs

typedef float v2f __attribute__((ext_vector_type(2)));
typedef float v8f __attribute__((ext_vector_type(8)));

// Order-preserving float <-> uint encoding so atomicMax(u32) == float max.
__device__ __forceinline__ unsigned enc_f32(float f) {
  unsigned u = __float_as_uint(f);
  return (u & 0x80000000u) ? ~u : (u | 0x80000000u);
}
__device__ __forceinline__ float dec_f32(unsigned u) {
  return (u & 0x80000000u) ? __uint_as_float(u & 0x7FFFFFFFu)
                           : __uint_as_float(~u);
}
#define ENC_NEG_INF 0x007FFFFFu  // enc(-inf)

__global__ void fill_u32(unsigned* __restrict__ p, unsigned v, int n) {
  int i = blockIdx.x * blockDim.x + threadIdx.x;
  if (i < n) p[i] = v;
}

// C[M,Nout] = A[M,K] @ W[K,Nout] + b, two weight sets selected by blockIdx.z.
// One wave32 per 16x32 output tile (two 16x16 WMMA accumulators sharing the
// A fetch). K swept 4/step with V_WMMA_F32_16X16X4_F32; main loop is fully
// branch-free (A = contiguous v2f -> b64 load), remainder handled once with
// clamped indices + value masking (no predicated loads).
// A layout: lane m=lane&15 holds A[m][k0+2*(lane>>4)+{0,1}] in {a.x,a.y}.
// B layout (mirror): lane n=lane&15 holds W[k0+2*(lane>>4)+{0,1}][n].
// D layout: vgpr r -> row m = r + 8*(lane>>4), col n = lane&15.
__global__ void gemm_wmma_f32(const float* __restrict__ A,
                              const float* __restrict__ W0, const float* __restrict__ b0,
                              float* __restrict__ O0,
                              const float* __restrict__ W1, const float* __restrict__ b1,
                              float* __restrict__ O1,
                              int K, int Nout) {
  const float* W = blockIdx.z ? W1 : W0;
  const float* b = blockIdx.z ? b1 : b0;
  float*       O = blockIdx.z ? O1 : O0;
  const int lane = threadIdx.x & 31;
  const int half = lane >> 4;
  const int lm   = lane & 15;
  const int mbase = blockIdx.y << 4;
  const int nbase = blockIdx.x << 5;       // 32 columns per wave

  v8f acc0 = {0.f, 0.f, 0.f, 0.f, 0.f, 0.f, 0.f, 0.f};
  v8f acc1 = acc0;
  const float* Arow  = A + (size_t)(mbase + lm) * K;
  const float* Bcol0 = W + nbase + lm;         // column n, first 16-wide tile
  const float* Bcol1 = Bcol0 + 16;             // second 16-wide tile

  int k0 = 0;
  for (; k0 + 4 <= K; k0 += 4) {
    const int ka = k0 + 2 * half;
    v2f av = *(const v2f*)(Arow + ka);         // contiguous, 8B aligned -> b64
    v2f bv0, bv1;
    bv0.x = Bcol0[(size_t)ka * Nout];
    bv0.y = Bcol0[(size_t)(ka + 1) * Nout];
    bv1.x = Bcol1[(size_t)ka * Nout];
    bv1.y = Bcol1[(size_t)(ka + 1) * Nout];
    acc0 = __builtin_amdgcn_wmma_f32_16x16x4_f32(false, av, false, bv0,
                                                 (short)0, acc0, false, false);
    acc1 = __builtin_amdgcn_wmma_f32_16x16x4_f32(false, av, false, bv1,
                                                 (short)0, acc1, false, false);
  }
  if (k0 < K) {                                // K remainder (layer 1: K=78, rem=2)
    const int ka  = k0 + 2 * half;
    const int kc0 = (ka     < K) ? ka     : (K - 1);   // clamped, always in-range
    const int kc1 = (ka + 1 < K) ? ka + 1 : (K - 1);
    const float m0 = (ka     < K) ? 1.0f : 0.0f;       // value mask -> cndmask
    const float m1 = (ka + 1 < K) ? 1.0f : 0.0f;
    v2f av, bv0, bv1;
    av.x  = Arow[kc0] * m0;
    av.y  = Arow[kc1] * m1;
    bv0.x = Bcol0[(size_t)kc0 * Nout] * m0;
    bv0.y = Bcol0[(size_t)kc1 * Nout] * m1;
    bv1.x = Bcol1[(size_t)kc0 * Nout] * m0;
    bv1.y = Bcol1[(size_t)kc1 * Nout] * m1;
    acc0 = __builtin_amdgcn_wmma_f32_16x16x4_f32(false, av, false, bv0,
                                                 (short)0, acc0, false, false);
    acc1 = __builtin_amdgcn_wmma_f32_16x16x4_f32(false, av, false, bv1,
                                                 (short)0, acc1, false, false);
  }

  const int   n0 = nbase + lm;
  const float bias0 = b[n0];
  const float bias1 = b[n0 + 16];
#pragma unroll
  for (int r = 0; r < 8; ++r) {
    int m = mbase + r + 8 * half;
    O[(size_t)m * Nout + n0]      = acc0[r] + bias0;
    O[(size_t)m * Nout + n0 + 16] = acc1[r] + bias1;
  }
}

// One wave per edge: logits[e,h] = att[h,:] . leaky(xl[src]+xr[dst]); atomic seg-max.
__global__ void edge_logits_kernel(const float* __restrict__ xl, const float* __restrict__ xr,
                                   const int* __restrict__ ei, const float* __restrict__ att,
                                   float* __restrict__ logits, unsigned* __restrict__ segmax,
                                   int Hh) {
  int wid  = (blockIdx.x * blockDim.x + threadIdx.x) >> 5;
  int lane = threadIdx.x & 31;
  if (wid >= ETOT) return;
  int s, d;
  if (wid < NEDGES) { s = ei[wid]; d = ei[NEDGES + wid]; }
  else              { s = wid - NEDGES; d = s; }
  const int HC = Hh * CCH;
  const float* pl = xl + (size_t)s * HC;
  const float* pr = xr + (size_t)d * HC;
  for (int h = 0; h < Hh; ++h) {
    float partial = 0.0f;
#pragma unroll
    for (int c = lane; c < CCH; c += 32) {
      float v = pl[h * CCH + c] + pr[h * CCH + c];
      v = (v > 0.0f) ? v : 0.2f * v;
      partial += att[h * CCH + c] * v;
    }
#pragma unroll
    for (int off = 16; off; off >>= 1)
      partial += __shfl_xor(partial, off, 32);
    if (lane == 0) {
      logits[(size_t)wid * Hh + h] = partial;
      atomicMax(&segmax[(size_t)d * Hh + h], enc_f32(partial));
    }
  }
}

// Thread per (edge, head): alpha = exp(logit - segmax[dst]); atomic seg-sum.
__global__ void edge_expsum_kernel(const int* __restrict__ ei, float* __restrict__ logits,
                                   const unsigned* __restrict__ segmax,
                                   float* __restrict__ segsum, int Hh) {
  int idx = blockIdx.x * blockDim.x + threadIdx.x;
  if (idx >= ETOT * Hh) return;
  int e = idx / Hh, h = idx - e * Hh;
  int d = (e < NEDGES) ? ei[NEDGES + e] : (e - NEDGES);
  float m = dec_f32(segmax[(size_t)d * Hh + h]);
  float a = expf(logits[idx] - m);
  logits[idx] = a;                       // reuse buffer: now holds exp values
  atomicAdd(&segsum[(size_t)d * Hh + h], a);
}

// One wave per edge: agg[dst] += (exp/segsum) * xl[src] via global_atomic_add_f32.
__global__ void edge_aggregate_kernel(const float* __restrict__ xl, const int* __restrict__ ei,
                                      const float* __restrict__ logits,
                                      const float* __restrict__ segsum,
                                      float* __restrict__ agg, int Hh) {
  int wid  = (blockIdx.x * blockDim.x + threadIdx.x) >> 5;
  int lane = threadIdx.x & 31;
  if (wid >= ETOT) return;
  int s, d;
  if (wid < NEDGES) { s = ei[wid]; d = ei[NEDGES + wid]; }
  else              { s = wid - NEDGES; d = s; }
  float wv[H1];
  for (int h = 0; h < Hh; ++h)
    wv[h] = logits[(size_t)wid * Hh + h] / segsum[(size_t)d * Hh + h];
  const int HC = Hh * CCH;
  const float* pl = xl + (size_t)s * HC;
  float*       pd = agg + (size_t)d * HC;
  for (int i = lane; i < HC; i += 32) {
    int h = i >> 7;                      // CCH == 128
    atomicAdd(&pd[i], wv[h] * pl[i]);
  }
}

__global__ void bias_elu_kernel(float* __restrict__ a, const float* __restrict__ bias,
                                int total, int F) {
  int i = blockIdx.x * blockDim.x + threadIdx.x;
  if (i >= total) return;
  float v = a[i] + bias[i % F];
  a[i] = (v > 0.0f) ? v : (expf(v) - 1.0f);
}

// ELU(agg2 + bias2) then atomic accumulate per graph; lane c==0 counts nodes.
__global__ void pool_accum_kernel(const float* __restrict__ agg2, const float* __restrict__ bias2,
                                  const int* __restrict__ batch,
                                  float* __restrict__ pool, float* __restrict__ cnt) {
  int i = blockIdx.x * blockDim.x + threadIdx.x;
  if (i >= NNODES * CCH) return;
  int n = i >> 7, c = i & 127;
  float v = agg2[i] + bias2[c];
  v = (v > 0.0f) ? v : (expf(v) - 1.0f);
  int g = batch[n];
  atomicAdd(&pool[(size_t)g * CCH + c], v);
  if (c == 0) atomicAdd(&cnt[g], 1.0f);
}

__global__ void pool_div_kernel(const float* __restrict__ pool, const float* __restrict__ cnt,
                                float* __restrict__ out) {
  int i = blockIdx.x * blockDim.x + threadIdx.x;
  if (i >= NG * CCH) return;
  out[i] = pool[i] / fmaxf(cnt[i >> 7], 1.0f);
}

extern "C" void kernel_launch(void* const* d_in, const int* in_sizes, int n_in,
                              void* d_out, int out_size, void* d_ws, size_t ws_size,
                              hipStream_t stream) {
  (void)in_sizes; (void)n_in; (void)out_size; (void)ws_size;
  const float* x     = (const float*)d_in[0];
  const int*   ei    = (const int*)d_in[1];
  const int*   batch = (const int*)d_in[2];
  const float* Wl1   = (const float*)d_in[3];
  const float* bl1   = (const float*)d_in[4];
  const float* Wr1   = (const float*)d_in[5];
  const float* br1   = (const float*)d_in[6];
  const float* att1  = (const float*)d_in[7];
  const float* bias1 = (const float*)d_in[8];
  const float* Wl2   = (const float*)d_in[9];
  const float* bl2   = (const float*)d_in[10];
  const float* Wr2   = (const float*)d_in[11];
  const float* br2   = (const float*)d_in[12];
  const float* att2  = (const float*)d_in[13];
  const float* bias2 = (const float*)d_in[14];
  float* out = (float*)d_out;

  // Carve workspace (~84 MB total; fits L2-resident working set)
  char* ws = (char*)d_ws;
  size_t off = 0;
  auto carve = [&](size_t bytes) -> void* {
    void* p = ws + off;
    off += (bytes + 255) & ~(size_t)255;
    return p;
  };
  float*    xl1     = (float*)carve((size_t)NNODES * H1 * CCH * 4);
  float*    xr1     = (float*)carve((size_t)NNODES * H1 * CCH * 4);
  float*    agg1    = (float*)carve((size_t)NNODES * H1 * CCH * 4);
  float*    logits1 = (float*)carve((size_t)ETOT * H1 * 4);
  unsigned* segmax1 = (unsigned*)carve((size_t)NNODES * H1 * 4);
  float*    segsum1 = (float*)carve((size_t)NNODES * H1 * 4);
  float*    xl2     = (float*)carve((size_t)NNODES * CCH * 4);
  float*    xr2     = (float*)carve((size_t)NNODES * CCH * 4);
  float*    agg2    = (float*)carve((size_t)NNODES * CCH * 4);
  float*    logits2 = (float*)carve((size_t)ETOT * 4);
  unsigned* segmax2 = (unsigned*)carve((size_t)NNODES * 4);
  float*    segsum2 = (float*)carve((size_t)NNODES * 4);
  float*    pool    = (float*)carve((size_t)NG * CCH * 4);
  float*    cnt     = (float*)carve((size_t)NG * 4);

  const int TB = 256;
  auto blocks = [](long long n, int tb) { return (unsigned)((n + tb - 1) / tb); };

  // Init: segment-max to enc(-inf), accumulators to zero (bit pattern 0 == 0.0f)
  fill_u32<<<blocks(NNODES * H1, TB), TB, 0, stream>>>(segmax1, ENC_NEG_INF, NNODES * H1);
  fill_u32<<<blocks(NNODES, TB), TB, 0, stream>>>(segmax2, ENC_NEG_INF, NNODES);
  fill_u32<<<blocks(NNODES * H1, TB), TB, 0, stream>>>((unsigned*)segsum1, 0u, NNODES * H1);
  fill_u32<<<blocks(NNODES, TB), TB, 0, stream>>>((unsigned*)segsum2, 0u, NNODES);
  fill_u32<<<blocks((long long)NNODES * H1 * CCH, TB), TB, 0, stream>>>((unsigned*)agg1, 0u, NNODES * H1 * CCH);
  fill_u32<<<blocks((long long)NNODES * CCH, TB), TB, 0, stream>>>((unsigned*)agg2, 0u, NNODES * CCH);
  fill_u32<<<blocks(NG * CCH, TB), TB, 0, stream>>>((unsigned*)pool, 0u, NG * CCH);
  fill_u32<<<blocks(NG, TB), TB, 0, stream>>>((unsigned*)cnt, 0u, NG);

  const unsigned edgeBlocks = blocks((long long)ETOT * 32, TB); // one wave per edge

  // ---- Layer 1 ----
  {
    dim3 grid(H1 * CCH / 32, NNODES / 16, 2);   // (16, 625, 2), 16x32 tile/wave
    gemm_wmma_f32<<<grid, 32, 0, stream>>>(x, Wl1, bl1, xl1, Wr1, br1, xr1, FIN, H1 * CCH);
  }
  edge_logits_kernel<<<edgeBlocks, TB, 0, stream>>>(xl1, xr1, ei, att1, logits1, segmax1, H1);
  edge_expsum_kernel<<<blocks((long long)ETOT * H1, TB), TB, 0, stream>>>(ei, logits1, segmax1, segsum1, H1);
  edge_aggregate_kernel<<<edgeBlocks, TB, 0, stream>>>(xl1, ei, logits1, segsum1, agg1, H1);
  bias_elu_kernel<<<blocks((long long)NNODES * H1 * CCH, TB), TB, 0, stream>>>(agg1, bias1, NNODES * H1 * CCH, H1 * CCH);

  // ---- Layer 2 ----
  {
    dim3 grid(CCH / 32, NNODES / 16, 2);        // (4, 625, 2)
    gemm_wmma_f32<<<grid, 32, 0, stream>>>(agg1, Wl2, bl2, xl2, Wr2, br2, xr2, H1 * CCH, CCH);
  }
  edge_logits_kernel<<<edgeBlocks, TB, 0, stream>>>(xl2, xr2, ei, att2, logits2, segmax2, 1);
  edge_expsum_kernel<<<blocks(ETOT, TB), TB, 0, stream>>>(ei, logits2, segmax2, segsum2, 1);
  edge_aggregate_kernel<<<edgeBlocks, TB, 0, stream>>>(xl2, ei, logits2, segsum2, agg2, 1);

  // ---- Pool ----
  pool_accum_kernel<<<blocks((long long)NNODES * CCH, TB), TB, 0, stream>>>(agg2, bias2, batch, pool, cnt);
  pool_div_kernel<<<blocks(NG * CCH, TB), TB, 0, stream>>>(pool, cnt, out);
}